// KGFM_73856257622617
// MI455X (gfx1250) — compile-verified
//
#include <hip/hip_runtime.h>
#include <hip/hip_bf16.h>

typedef __attribute__((ext_vector_type(2))) float v2f;
typedef __attribute__((ext_vector_type(8))) float v8f;

#define D2 128
#define KNBR 64

// ---------------------------------------------------------------------------
// CDNA5 async global->LDS copy (ASYNCcnt-tracked), 16 bytes per lane.
// vdst = LDS byte offset (low 32 bits of generic shared pointer),
// vaddr = 64-bit global address.  Portable inline-asm form (bypasses the
// toolchain-divergent tensor builtins).
// ---------------------------------------------------------------------------
__device__ __forceinline__ void async_ld16(void* lds, const void* gptr)
{
    uint32_t off = (uint32_t)(uintptr_t)lds;   // low 32 bits = LDS offset
    asm volatile("global_load_async_to_lds_b128 %0, %1, off"
                 :: "v"(off), "v"(gptr) : "memory");
}
__device__ __forceinline__ void wait_async0()
{
    asm volatile("s_wait_asynccnt 0x0" ::: "memory");
}

// ---------------------------------------------------------------------------
// Kernel 1: h[b2][d] = bn(entity_embs[idx[b2]][d]); rows 0..B-1 use i[], rows
// B..2B-1 use u[].  bn(x) = (x-mean)*gamma*rsqrt(var+eps)+beta  (affine).
// ---------------------------------------------------------------------------
__global__ __launch_bounds__(256)
void bn_gather(const int* __restrict__ iIdx, const int* __restrict__ uIdx,
               const float* __restrict__ E,
               const float* __restrict__ gamma, const float* __restrict__ beta,
               const float* __restrict__ mean,  const float* __restrict__ var,
               float* __restrict__ h, int B)
{
    int t = blockIdx.x * blockDim.x + threadIdx.x;
    if (t >= 2 * B * D2) return;
    int d  = t & (D2 - 1);
    int b2 = t >> 7;
    int idx = (b2 < B) ? iIdx[b2] : uIdx[b2 - B];
    float scale = gamma[d] * rsqrtf(var[d] + 1e-5f);
    float x = E[(size_t)idx * D2 + d];
    h[(size_t)b2 * D2 + d] = (x - mean[d]) * scale + beta[d];
}

// ---------------------------------------------------------------------------
// Kernel 2 (generic): C[M,128] = epilogue(A[M,128] @ W[128,128] + bias)
//   act==1 : leaky-relu(0.2);   resid : C = resid + activated(...)
// Block = 256 threads (8 waves) -> one 64-row x 128-col tile.
// Wave w: row-tile rt = w&3 (16 rows), col-half ch = w>>2 (4 x 16 cols).
// Both W (128x132 padded, 66KB) and the A tile (64x132, 33KB) are staged in
// dynamic LDS via GLOBAL_LOAD_ASYNC_TO_LDS_B128; stride 132 kills the
// stride-128 bank conflicts on column reads and row-pair reads.
// ---------------------------------------------------------------------------
#define TSTRIDE 132
#define WS_FLOATS (D2 * TSTRIDE)           // 16896 floats
#define AS_FLOATS (64 * TSTRIDE)           // 8448 floats
#define GEMM_SMEM_BYTES ((WS_FLOATS + AS_FLOATS) * 4)

__global__ __launch_bounds__(256)
void gemm128_wmma(const float* __restrict__ A, const float* __restrict__ W,
                  const float* __restrict__ bias, const float* __restrict__ resid,
                  float* __restrict__ C, int M, int act)
{
    extern __shared__ float smem[];
    float* Ws = smem;                      // [128][132]
    float* As = smem + WS_FLOATS;          // [64][132]

    const int tid     = threadIdx.x;
    const int rowBase = blockIdx.x * 64;

    // Async-stage W: 4096 16B chunks; chunk c -> row c/32, col-chunk c%32.
    for (int c = tid; c < D2 * (D2 / 4); c += 256) {
        int r  = c >> 5;
        int cc = c & 31;
        async_ld16(&Ws[r * TSTRIDE + cc * 4], W + r * D2 + cc * 4);
    }
    // Async-stage A tile: 2048 16B chunks (guard rows for the last block).
    for (int c = tid; c < 64 * (D2 / 4); c += 256) {
        int r  = c >> 5;
        int cc = c & 31;
        if (rowBase + r < M)
            async_ld16(&As[r * TSTRIDE + cc * 4],
                       A + (size_t)(rowBase + r) * D2 + cc * 4);
    }
    wait_async0();
    __syncthreads();

    const int wave  = tid >> 5;
    const int lane  = tid & 31;
    const int rt    = wave & 3;        // row tile 0..3
    const int ch    = wave >> 2;       // col half 0..1
    const int lrow  = lane & 15;       // M (A) or N (B,C) sub-index
    const int khalf = lane >> 4;       // 0/1 -> K offset 0/2

    v8f acc[4] = {v8f{}, v8f{}, v8f{}, v8f{}};

    const float* Asub = As + (rt * 16 + lrow) * TSTRIDE;
    const float* Wsub = Ws + ch * 64 + lrow;

    for (int k = 0; k < D2; k += 4) {
        // A fragment: row M=lrow, K = k + 2*khalf + {0,1}
        v2f a;
        a.x = Asub[k + 2 * khalf + 0];
        a.y = Asub[k + 2 * khalf + 1];
        // B fragments: W[k + 2*khalf + {0,1}][ch*64 + 16*t + lrow]
        const float* w0 = Wsub + (k + 2 * khalf) * TSTRIDE;
        v2f b0; b0.x = w0[0];  b0.y = w0[TSTRIDE + 0];
        v2f b1; b1.x = w0[16]; b1.y = w0[TSTRIDE + 16];
        v2f b2; b2.x = w0[32]; b2.y = w0[TSTRIDE + 32];
        v2f b3; b3.x = w0[48]; b3.y = w0[TSTRIDE + 48];
        acc[0] = __builtin_amdgcn_wmma_f32_16x16x4_f32(false, a, false, b0,
                                                       (short)0, acc[0], false, false);
        acc[1] = __builtin_amdgcn_wmma_f32_16x16x4_f32(false, a, false, b1,
                                                       (short)0, acc[1], false, false);
        acc[2] = __builtin_amdgcn_wmma_f32_16x16x4_f32(false, a, false, b2,
                                                       (short)0, acc[2], false, false);
        acc[3] = __builtin_amdgcn_wmma_f32_16x16x4_f32(false, a, false, b3,
                                                       (short)0, acc[3], false, false);
    }

    // Epilogue. C layout: vgpr v -> M = v + 8*khalf (local), N = lane%16.
    const int mBase = rowBase + rt * 16 + 8 * khalf;
#pragma unroll
    for (int t = 0; t < 4; ++t) {
        int ncol = ch * 64 + 16 * t + lrow;
        float bv = bias[ncol];
#pragma unroll
        for (int v = 0; v < 8; ++v) {
            int row = mBase + v;
            if (row < M) {
                float val = acc[t][v] + bv;
                if (act) val = (val >= 0.f) ? val : 0.2f * val;
                if (resid) val += resid[(size_t)row * D2 + ncol];
                C[(size_t)row * D2 + ncol] = val;
            }
        }
    }
}

// ---------------------------------------------------------------------------
// Kernel 3: FM aggregation. One block per (batch element, layer):
//   for k in 0..63:  tr = Ewr[adjE[idx][k]]       (gather, L2-resident)
//                    hr = tanh(hWr[b2] + R[adjR[idx][k]])
//                    x  = hr*tr;  s += x; ss += x*x
//   A2[b2] = h[b2] + (s*s - ss)
// Relation table (16 KB) async-staged into LDS.
// ---------------------------------------------------------------------------
__global__ __launch_bounds__(128)
void fm_layer(const int* __restrict__ iIdx, const int* __restrict__ uIdx,
              const int* __restrict__ adjE, const int* __restrict__ adjR,
              const float* __restrict__ Ewr, const float* __restrict__ R,
              const float* __restrict__ hWr, const float* __restrict__ h,
              float* __restrict__ A2, int B)
{
    __shared__ float relS[32 * D2];    // full relation table, 16 KB
    __shared__ int   eIds[KNBR];
    __shared__ int   rIds[KNBR];

    const int b2 = blockIdx.x;         // 0..2B-1
    const int d  = threadIdx.x;        // 0..127

    // Async-stage relation table: 1024 16B chunks / 128 threads = 8 each.
    for (int c = d; c < 32 * (D2 / 4); c += 128)
        async_ld16(&relS[c * 4], R + c * 4);

    const int idx = (b2 < B) ? iIdx[b2] : uIdx[b2 - B];
    if (d < KNBR) {
        eIds[d] = adjE[(size_t)idx * KNBR + d];
        rIds[d] = adjR[(size_t)idx * KNBR + d];
    }
    wait_async0();
    __syncthreads();

    const float hw = hWr[(size_t)b2 * D2 + d];
    float s = 0.f, ss = 0.f;
#pragma unroll 4
    for (int k = 0; k < KNBR; ++k) {
        if (k + 8 < KNBR)
            __builtin_prefetch(&Ewr[(size_t)eIds[k + 8] * D2 + d], 0, 1);
        float tr = Ewr[(size_t)eIds[k] * D2 + d];
        float hr = tanhf(hw + relS[rIds[k] * D2 + d]);
        float x  = hr * tr;
        s  += x;
        ss += x * x;
    }
    float Nh = s * s - ss;
    A2[(size_t)b2 * D2 + d] = h[(size_t)b2 * D2 + d] + Nh;
}

// ---------------------------------------------------------------------------
// Kernel 4: out[b] = dot(concat(u_e1[b], h1_i[b]), wl_w) + wl_b
// h1 rows 0..B-1 = i-layer, rows B..2B-1 = u-layer.
// ---------------------------------------------------------------------------
__global__ __launch_bounds__(128)
void final_dot(const float* __restrict__ h1, const float* __restrict__ wl_w,
               const float* __restrict__ wl_b, float* __restrict__ out, int B)
{
    __shared__ float red[D2];
    const int b = blockIdx.x;
    const int d = threadIdx.x;
    float v = h1[(size_t)(B + b) * D2 + d] * wl_w[d]          // u_e part
            + h1[(size_t)b * D2 + d]       * wl_w[D2 + d];    // h part
    red[d] = v;
    __syncthreads();
    for (int s = 64; s > 0; s >>= 1) {
        if (d < s) red[d] += red[d + s];
        __syncthreads();
    }
    if (d == 0) out[b] = red[0] + wl_b[0];
}

// ---------------------------------------------------------------------------
extern "C" void kernel_launch(void* const* d_in, const int* in_sizes, int n_in,
                              void* d_out, int out_size, void* d_ws, size_t ws_size,
                              hipStream_t stream)
{
    const int*   u     = (const int*)d_in[0];
    const int*   iI    = (const int*)d_in[1];
    const int*   adjE  = (const int*)d_in[2];
    const int*   adjR  = (const int*)d_in[3];
    const float* E     = (const float*)d_in[4];
    const float* R     = (const float*)d_in[5];
    const float* gamma = (const float*)d_in[6];
    const float* beta  = (const float*)d_in[7];
    const float* mean  = (const float*)d_in[8];
    const float* var   = (const float*)d_in[9];
    const float* Wr1_w = (const float*)d_in[10];
    const float* Wr1_b = (const float*)d_in[11];
    const float* W1_w  = (const float*)d_in[12];
    const float* W1_b  = (const float*)d_in[13];
    const float* wl_w  = (const float*)d_in[14];
    const float* wl_b  = (const float*)d_in[15];
    float* out = (float*)d_out;

    const int B    = in_sizes[0];
    const int nEnt = in_sizes[4] / D2;
    const int B2   = 2 * B;

    // Workspace layout (floats):
    float* ws  = (float*)d_ws;
    float* Ewr = ws;                               // [nEnt,128]  = E @ Wr1 + b
    float* h   = Ewr + (size_t)nEnt * D2;          // [2B,128] bn'd gathered rows
    float* hWr = h   + (size_t)B2 * D2;            // [2B,128]
    float* A2  = hWr + (size_t)B2 * D2;            // [2B,128] h + Nh
    float* h1  = A2  + (size_t)B2 * D2;            // [2B,128] layer output

    // 1) h = bn(E[idx])
    {
        int total = B2 * D2;
        bn_gather<<<(total + 255) / 256, 256, 0, stream>>>(
            iI, u, E, gamma, beta, mean, var, h, B);
    }
    // 2) Ewr = E @ Wr1 + b    (shared by both layers; B*K > nEnt so this is
    //    strictly cheaper than per-neighbor GEMM)
    gemm128_wmma<<<(nEnt + 63) / 64, 256, GEMM_SMEM_BYTES, stream>>>(
        E, Wr1_w, Wr1_b, nullptr, Ewr, nEnt, 0);
    // 3) hWr = h @ Wr1 + b    (both layers in one launch)
    gemm128_wmma<<<(B2 + 63) / 64, 256, GEMM_SMEM_BYTES, stream>>>(
        h, Wr1_w, Wr1_b, nullptr, hWr, B2, 0);
    // 4) A2 = h + Nh (FM aggregation over 64 neighbors)
    fm_layer<<<B2, 128, 0, stream>>>(iI, u, adjE, adjR, Ewr, R, hWr, h, A2, B);
    // 5) h1 = h + lrelu(A2 @ W1 + b)
    gemm128_wmma<<<(B2 + 63) / 64, 256, GEMM_SMEM_BYTES, stream>>>(
        A2, W1_w, W1_b, h, h1, B2, 1);
    // 6) out = concat(u_e1, h1_i) @ wl_w + wl_b
    final_dot<<<B, 128, 0, stream>>>(h1, wl_w, wl_b, out, B);
}